// DeepseekV3Layer_42485816492040
// MI455X (gfx1250) — compile-verified
//
#include <hip/hip_runtime.h>
#include <hip/hip_bf16.h>

typedef __attribute__((ext_vector_type(2))) float v2f;
typedef __attribute__((ext_vector_type(8))) float v8f;

// Problem constants (match reference)
#define B_    32
#define S_    4096
#define S1    4097
#define HID   4096
#define H_    32
#define NOPE  128
#define ROPE_ 64
#define HEAD  192
#define V_    128
#define KVL   512
#define QL    1536
#define INTER 11008
#define EPS_  1e-6f
#define SCALE_ 0.07216878364870323f   // 192^-0.5

// ---------------------------------------------------------------------------
// Generic WMMA f32 GEMM: D[z][m][n] = alpha * sum_k A[z][m][k] * B[z][k][n]
//                                      (+ E[z][m][n] if E != null)
// One wave computes a 16x64 tile of D (4 WMMA 16x16 column tiles): the A
// fragment is loaded once per k-step and reused by 4 V_WMMA_F32_16X16X4_F32
// ops.  All B addresses derive from ONE running pointer (chained directly
// off the kernel argument) plus loop-invariant column offsets, so LLVM's
// infer-address-spaces keeps the loads on the GLOBAL path (global_load_*,
// LOADcnt only) instead of falling back to FLAT.
// blockIdx = (nTile64, mTile16, z).  N must be a multiple of 64.
//
// Fragment layouts per CDNA5 ISA 7.12.2 (32-bit, wave32):
//   A 16x4 : lane 0-15 -> M=lane, VGPR0 K={0|2}, VGPR1 K={1|3} by lane half
//   B 4x16 : VGPRj lanes0-15 -> K=j, lanes16-31 -> K=2+j, N=lane&15
//   C/D    : VGPRi lanes0-15 -> M=i,  lanes16-31 -> M=8+i, N=lane&15
// ---------------------------------------------------------------------------
#define NT 4   // 16-wide column tiles per wave

__global__ __launch_bounds__(32) void gemm_wmma_f32(
    const float* __restrict__ A, long sAm, long aB,
    const float* __restrict__ Bm, long sBk, long sBn, long bB,
    float* __restrict__ D, long sDm, long sDn, long dB,
    const float* __restrict__ E,
    int K, float alpha)
{
  const int lane = threadIdx.x;
  const int rc   = lane & 15;            // row (A) / col (B,D) within tile
  const int hi   = lane >> 4;            // 0: lanes 0-15, 1: lanes 16-31
  const int koff = hi << 1;              // K sub-offset for A/B fragments
  const int row0 = blockIdx.y * 16;
  const int col0 = blockIdx.x * (16 * NT);
  const long z   = blockIdx.z;

  const float* __restrict__ Arow = A + z * aB + (long)(row0 + rc) * sAm + koff;

  // Single running B pointer; column tiles are fixed offsets from it.
  const float* __restrict__ Bp =
      Bm + z * bB + (long)(col0 + rc) * sBn + (long)koff * sBk;
  const long n1 = 16 * sBn;          // column-tile offsets
  const long n2 = 32 * sBn;
  const long n3 = 48 * sBn;
  const long stepB = 4 * sBk;

  v8f c0 = {0.f,0.f,0.f,0.f,0.f,0.f,0.f,0.f};
  v8f c1 = c0, c2 = c0, c3 = c0;

#pragma unroll 2
  for (int k0 = 0; k0 < K; k0 += 4) {
    v2f a;
    a[0] = Arow[k0];          // K = k0 + koff
    a[1] = Arow[k0 + 1];      // K = k0 + koff + 1
    __builtin_prefetch(Arow + k0 + 32, 0, 1);
    __builtin_prefetch(Bp + 8 * stepB, 0, 1);

    v2f b;
    b[0] = Bp[0];       b[1] = Bp[sBk];
    c0 = __builtin_amdgcn_wmma_f32_16x16x4_f32(false, a, false, b,
                                               (short)0, c0, false, false);
    b[0] = Bp[n1];      b[1] = Bp[n1 + sBk];
    c1 = __builtin_amdgcn_wmma_f32_16x16x4_f32(false, a, false, b,
                                               (short)0, c1, false, false);
    b[0] = Bp[n2];      b[1] = Bp[n2 + sBk];
    c2 = __builtin_amdgcn_wmma_f32_16x16x4_f32(false, a, false, b,
                                               (short)0, c2, false, false);
    b[0] = Bp[n3];      b[1] = Bp[n3 + sBk];
    c3 = __builtin_amdgcn_wmma_f32_16x16x4_f32(false, a, false, b,
                                               (short)0, c3, false, false);
    Bp += stepB;
  }

  v8f acc[NT] = {c0, c1, c2, c3};
#pragma unroll
  for (int t = 0; t < NT; ++t) {
#pragma unroll
    for (int i = 0; i < 8; ++i) {
      const int  m   = i + hi * 8;
      const long off = z * dB + (long)(row0 + m) * sDm
                              + (long)(col0 + t * 16 + rc) * sDn;
      float v = alpha * acc[t][i];
      if (E) v += E[off];
      D[off] = v;
    }
  }
}

// ---------------------------------------------------------------------------
// Elementwise / reduction helpers
// ---------------------------------------------------------------------------
__device__ __forceinline__ float block_reduce_sum(float v, float* red) {
  red[threadIdx.x] = v;
  __syncthreads();
  for (int off = blockDim.x >> 1; off > 0; off >>= 1) {
    if ((int)threadIdx.x < off) red[threadIdx.x] += red[threadIdx.x + off];
    __syncthreads();
  }
  float r = red[0];
  __syncthreads();
  return r;
}

__device__ __forceinline__ float block_reduce_max(float v, float* red) {
  red[threadIdx.x] = v;
  __syncthreads();
  for (int off = blockDim.x >> 1; off > 0; off >>= 1) {
    if ((int)threadIdx.x < off)
      red[threadIdx.x] = fmaxf(red[threadIdx.x], red[threadIdx.x + off]);
    __syncthreads();
  }
  float r = red[0];
  __syncthreads();
  return r;
}

// RMSNorm: one block per row.  Works in-place (each element read+written by
// the same thread).
__global__ void rmsnorm_kernel(const float* __restrict__ x,
                               const float* __restrict__ w,
                               float* __restrict__ out, int n) {
  __shared__ float red[256];
  const long row = blockIdx.x;
  const float* xr = x + row * n;
  float s = 0.f;
  for (int i = threadIdx.x; i < n; i += blockDim.x) { float v = xr[i]; s += v * v; }
  s = block_reduce_sum(s, red);
  const float rs = rsqrtf(s / (float)n + EPS_);
  for (int i = threadIdx.x; i < n; i += blockDim.x)
    out[row * n + i] = xr[i] * rs * w[i];
}

// Per-batch: RMSNorm kv_c -> latent[:512]; RoPE(k_pe) -> latent[512:576];
// RoPE(q_pe per head) -> q_full[..., 512:576].
__global__ void prep_kernel(const float* __restrict__ ckv,
                            const float* __restrict__ kv_ln_w,
                            const float* __restrict__ cos_t,
                            const float* __restrict__ sin_t,
                            const float* __restrict__ q,
                            float* __restrict__ latent,
                            float* __restrict__ q_full) {
  __shared__ float red[128];
  const int b = blockIdx.x;
  const int tid = threadIdx.x;
  const float* crow = ckv + (long)b * (KVL + ROPE_);
  float* lrow = latent + (long)b * (KVL + ROPE_);

  float s = 0.f;
  for (int i = tid; i < KVL; i += 128) { float v = crow[i]; s += v * v; }
  s = block_reduce_sum(s, red);
  const float rs = rsqrtf(s / (float)KVL + EPS_);
  for (int i = tid; i < KVL; i += 128) lrow[i] = crow[i] * rs * kv_ln_w[i];

  const float* cs = cos_t + (long)b * (ROPE_ / 2);
  const float* sn = sin_t + (long)b * (ROPE_ / 2);

  // k_pe rope: out[i] = x[2i]*c - x[2i+1]*s ; out[32+i] = x[2i+1]*c + x[2i]*s
  for (int i = tid; i < ROPE_ / 2; i += 128) {
    const float x1 = crow[KVL + 2 * i], x2 = crow[KVL + 2 * i + 1];
    lrow[KVL + i]              = x1 * cs[i] - x2 * sn[i];
    lrow[KVL + ROPE_ / 2 + i]  = x2 * cs[i] + x1 * sn[i];
  }

  // q_pe rope for all heads
  for (int t = tid; t < H_ * (ROPE_ / 2); t += 128) {
    const int h = t >> 5, i = t & 31;
    const float* qp = q + ((long)b * H_ + h) * HEAD + NOPE;
    float* dst = q_full + ((long)b * H_ + h) * (KVL + ROPE_) + KVL;
    const float x1 = qp[2 * i], x2 = qp[2 * i + 1];
    dst[i]              = x1 * cs[i] - x2 * sn[i];
    dst[ROPE_ / 2 + i]  = x2 * cs[i] + x1 * sn[i];
  }
}

// One block per (b,h): append the latent-position score, then softmax over
// the full 4097-length row (in place inside attn buffer).
__global__ void softmax_kernel(const float* __restrict__ q_full,
                               const float* __restrict__ latent,
                               float* __restrict__ attn) {
  __shared__ float red[256];
  const long bh = blockIdx.x;
  const long b  = bh >> 5;
  const float* qr  = q_full + bh * (KVL + ROPE_);
  const float* lat = latent + b * (KVL + ROPE_);
  float* arow = attn + bh * S1;
  const int tid = threadIdx.x;

  float s = 0.f;
  for (int i = tid; i < KVL + ROPE_; i += 256) s += qr[i] * lat[i];
  s = block_reduce_sum(s, red);
  if (tid == 0) arow[S_] = s * SCALE_;
  __syncthreads();

  float m = -3.0e38f;
  for (int i = tid; i < S1; i += 256) m = fmaxf(m, arow[i]);
  m = block_reduce_max(m, red);

  float sum = 0.f;
  for (int i = tid; i < S1; i += 256) {
    const float e = __expf(arow[i] - m);
    arow[i] = e;
    sum += e;
  }
  sum = block_reduce_sum(sum, red);
  const float inv = 1.f / sum;
  for (int i = tid; i < S1; i += 256) arow[i] *= inv;
}

// o_lat[b,h,l] += attn[b,h,4096] * latent[b,l]  (the appended KV position)
__global__ void olat_tail_kernel(const float* __restrict__ attn,
                                 const float* __restrict__ latent,
                                 float* __restrict__ o_lat) {
  const long idx = (long)blockIdx.x * blockDim.x + threadIdx.x;  // B*H*512
  const int  l  = (int)(idx & (KVL - 1));
  const long bh = idx >> 9;
  const long b  = bh >> 5;
  o_lat[idx] += attn[bh * S1 + S_] * latent[b * (KVL + ROPE_) + l];
}

// act[b,i] = silu(gate) * up, gate = gu[b, i], up = gu[b, INTER + i]
__global__ void silu_kernel(const float* __restrict__ gu,
                            float* __restrict__ act) {
  const long idx = (long)blockIdx.x * blockDim.x + threadIdx.x;  // B*INTER
  const long b = idx / INTER;
  const int  i = (int)(idx - b * INTER);
  const float g = gu[b * (2 * INTER) + i];
  const float u = gu[b * (2 * INTER) + INTER + i];
  act[idx] = (g / (1.f + __expf(-g))) * u;
}

// ---------------------------------------------------------------------------
// Host side
// ---------------------------------------------------------------------------
extern "C" void kernel_launch(void* const* d_in, const int* in_sizes, int n_in,
                              void* d_out, int out_size, void* d_ws, size_t ws_size,
                              hipStream_t stream) {
  const float* hidden   = (const float*)d_in[0];
  const float* cos_t    = (const float*)d_in[1];
  const float* sin_t    = (const float*)d_in[2];
  const float* kv_cache = (const float*)d_in[3];
  const float* w_q_a    = (const float*)d_in[4];
  const float* q_a_ln_w = (const float*)d_in[5];
  const float* w_q_b    = (const float*)d_in[6];
  const float* w_kv_a   = (const float*)d_in[7];
  const float* kv_ln_w  = (const float*)d_in[8];
  const float* w_kv_b   = (const float*)d_in[9];
  const float* w_o      = (const float*)d_in[10];
  const float* in_ln_w  = (const float*)d_in[11];
  const float* post_ln_w= (const float*)d_in[12];
  const float* w_gu     = (const float*)d_in[13];
  const float* w_down   = (const float*)d_in[14];
  float* out = (float*)d_out;

  float* ws = (float*)d_ws;
  float* x_norm = ws;                         // 32*4096   = 131072
  float* q_c    = x_norm + 131072;            // 32*1536   = 49152
  float* q      = q_c    + 49152;             // 32*6144   = 196608
  float* ckv    = q      + 196608;            // 32*576    = 18432
  float* latent = ckv    + 18432;             // 32*576    = 18432
  float* q_full = latent + 18432;             // 32*32*576 = 589824
  float* attn   = q_full + 589824;            // 32*32*4097= 4195328
  float* o_lat  = attn   + 4196352;           // 32*32*512 = 524288
  float* obuf   = o_lat  + 524288;            // 32*4096   = 131072
  float* hbuf   = obuf   + 131072;            // 32*4096   = 131072
  float* x2     = hbuf   + 131072;            // 32*4096   = 131072
  float* gu     = x2     + 131072;            // 32*22016  = 704512
  float* act    = gu     + 704512;            // 32*11008  = 352256
  (void)ws_size; (void)n_in; (void)in_sizes; (void)out_size;

  const long KV_B = (long)S_ * (KVL + ROPE_);   // 2359296: kv_cache batch stride
  const long QF   = KVL + ROPE_;                // 576
  const long AT_B = (long)H_ * S1;              // 131104

  // 1. x = rms(hidden, input_ln_w)
  hipLaunchKernelGGL(rmsnorm_kernel, dim3(B_), dim3(256), 0, stream,
                     hidden, in_ln_w, x_norm, HID);
  // 2. q_c = x @ w_q_a  (32x4096 @ 4096x1536)
  hipLaunchKernelGGL(gemm_wmma_f32, dim3(QL / 64, 2, 1), dim3(32), 0, stream,
                     x_norm, (long)HID, 0L, w_q_a, (long)QL, 1L, 0L,
                     q_c, (long)QL, 1L, 0L, (const float*)nullptr, HID, 1.f);
  // 3. q_c = rms(q_c, q_a_ln_w)
  hipLaunchKernelGGL(rmsnorm_kernel, dim3(B_), dim3(256), 0, stream,
                     q_c, q_a_ln_w, q_c, QL);
  // 4. q = q_c @ w_q_b  (32x1536 @ 1536x6144)
  hipLaunchKernelGGL(gemm_wmma_f32, dim3(H_ * HEAD / 64, 2, 1), dim3(32), 0, stream,
                     q_c, (long)QL, 0L, w_q_b, (long)(H_ * HEAD), 1L, 0L,
                     q, (long)(H_ * HEAD), 1L, 0L, (const float*)nullptr, QL, 1.f);
  // 5. ckv = x @ w_kv_a  (32x4096 @ 4096x576)
  hipLaunchKernelGGL(gemm_wmma_f32, dim3((int)QF / 64, 2, 1), dim3(32), 0, stream,
                     x_norm, (long)HID, 0L, w_kv_a, QF, 1L, 0L,
                     ckv, QF, 1L, 0L, (const float*)nullptr, HID, 1.f);
  // 6. latent = [rms(kv_c), rope(k_pe)]; q_full[...,512:] = rope(q_pe)
  hipLaunchKernelGGL(prep_kernel, dim3(B_), dim3(128), 0, stream,
                     ckv, kv_ln_w, cos_t, sin_t, q, latent, q_full);
  // 7. q_full[...,:512] = einsum('bhn,lhn->bhl', q_nope, W_UK)  (per head z)
  hipLaunchKernelGGL(gemm_wmma_f32, dim3(KVL / 64, 2, H_), dim3(32), 0, stream,
                     q, (long)(H_ * HEAD), (long)HEAD,
                     w_kv_b, 1L, (long)(H_ * (NOPE + V_)), (long)(NOPE + V_),
                     q_full, (long)(H_ * QF), 1L, QF,
                     (const float*)nullptr, NOPE, 1.f);
  // 8. attn[...,:4096] = SCALE * q_full @ kv_cache^T  (per batch z)
  hipLaunchKernelGGL(gemm_wmma_f32, dim3(S_ / 64, 2, B_), dim3(32), 0, stream,
                     q_full, QF, (long)(H_ * QF),
                     kv_cache, 1L, QF, KV_B,
                     attn, (long)S1, 1L, AT_B,
                     (const float*)nullptr, (int)QF, SCALE_);
  // 9. appended-position score + softmax over 4097 (per (b,h))
  hipLaunchKernelGGL(softmax_kernel, dim3(B_ * H_), dim3(256), 0, stream,
                     q_full, latent, attn);
  // 10. o_lat = attn[:,:,:4096] @ kv_cache[...,:512]  (per batch z)
  hipLaunchKernelGGL(gemm_wmma_f32, dim3(KVL / 64, 2, B_), dim3(32), 0, stream,
                     attn, (long)S1, AT_B,
                     kv_cache, QF, 1L, KV_B,
                     o_lat, (long)KVL, 1L, (long)(H_ * KVL),
                     (const float*)nullptr, S_, 1.f);
  // 11. o_lat += attn[...,4096] * latent[...,:512]
  hipLaunchKernelGGL(olat_tail_kernel, dim3(B_ * H_ * KVL / 256), dim3(256), 0, stream,
                     attn, latent, o_lat);
  // 12. o = einsum('bhl,lhv->bhv', o_lat, W_UV)  (per head z)
  hipLaunchKernelGGL(gemm_wmma_f32, dim3(V_ / 64, 2, H_), dim3(32), 0, stream,
                     o_lat, (long)(H_ * KVL), (long)KVL,
                     w_kv_b + NOPE, (long)(H_ * (NOPE + V_)), 1L, (long)(NOPE + V_),
                     obuf, (long)HID, 1L, (long)V_,
                     (const float*)nullptr, KVL, 1.f);
  // 13. h = hidden + o @ w_o
  hipLaunchKernelGGL(gemm_wmma_f32, dim3(HID / 64, 2, 1), dim3(32), 0, stream,
                     obuf, (long)HID, 0L, w_o, (long)HID, 1L, 0L,
                     hbuf, (long)HID, 1L, 0L, hidden, H_ * V_, 1.f);
  // 14. x2 = rms(h, post_ln_w)
  hipLaunchKernelGGL(rmsnorm_kernel, dim3(B_), dim3(256), 0, stream,
                     hbuf, post_ln_w, x2, HID);
  // 15. gu = x2 @ w_gate_up  (32x4096 @ 4096x22016)
  hipLaunchKernelGGL(gemm_wmma_f32, dim3(2 * INTER / 64, 2, 1), dim3(32), 0, stream,
                     x2, (long)HID, 0L, w_gu, (long)(2 * INTER), 1L, 0L,
                     gu, (long)(2 * INTER), 1L, 0L, (const float*)nullptr, HID, 1.f);
  // 16. act = silu(gate) * up
  hipLaunchKernelGGL(silu_kernel, dim3(B_ * INTER / 256), dim3(256), 0, stream,
                     gu, act);
  // 17. out = h + act @ w_down
  hipLaunchKernelGGL(gemm_wmma_f32, dim3(HID / 64, 2, 1), dim3(32), 0, stream,
                     act, (long)INTER, 0L, w_down, (long)HID, 1L, 0L,
                     out, (long)HID, 1L, 0L, hbuf, INTER, 1.f);
}